// VertexHybridModel_15272903705028
// MI455X (gfx1250) — compile-verified
//
#include <hip/hip_runtime.h>
#include <math.h>

// ---------------------------------------------------------------------------
// VertexHybridModel forward for MI455X (gfx1250), fp32 end-to-end.
// Dense linears use V_WMMA_F32_16X16X4_F32 (native fp32 WMMA, wave32).
// Attention is algebraically reduced: out = alpha@V + S@wev + bev,
// with S_i = sum_j alpha_ij * relu(Q_i + K_j).
// ---------------------------------------------------------------------------

typedef __attribute__((ext_vector_type(2))) float v2f;
typedef __attribute__((ext_vector_type(8))) float v8f;

constexpr int NN      = 1024;   // nodes
constexpr int H       = 128;    // hidden
constexpr int OUT_DIM = 64;
constexpr int EDGES   = 32768;

// ---------------------------------------------------------------------------
// WMMA fp32 GEMM: C[M,N] = act(A[M,K] @ B[K,N] + bias (+ add0 + add1))
// One wave computes one 16x16 tile; K stepped by 4 via v_wmma_f32_16x16x4_f32.
// Tile index is wave-uniform so EXEC stays all-ones for WMMA.
// ---------------------------------------------------------------------------
template <bool RELU>
__global__ void wmma_gemm_f32(const float* __restrict__ A, const float* __restrict__ B,
                              const float* __restrict__ bias,
                              const float* __restrict__ add0, const float* __restrict__ add1,
                              float* __restrict__ C, int M, int N, int K) {
    const int wavesPerBlock = blockDim.x >> 5;
    const int wid   = blockIdx.x * wavesPerBlock + (threadIdx.x >> 5);
    const int tilesN = N >> 4;
    const int numTiles = (M >> 4) * tilesN;
    if (wid >= numTiles) return;                 // wave-uniform branch

    const int tm = wid / tilesN;
    const int tn = wid - tm * tilesN;
    const int lane = threadIdx.x & 31;
    const int half = lane >> 4;                  // 0 or 1
    const int idx  = lane & 15;
    const int m0 = tm << 4, n0 = tn << 4;

    v8f acc = {0.f, 0.f, 0.f, 0.f, 0.f, 0.f, 0.f, 0.f};

    // A fragment: lane(idx) = row m0+idx; VGPR pair holds K = {2h, 2h+1} (+k)
    const float* arow = A + (size_t)(m0 + idx) * K + 2 * half;
    // B fragment: lane(idx) = col n0+idx; same K striping
    const float* bcol = B + (size_t)(2 * half) * N + (n0 + idx);

    for (int k = 0; k < K; k += 4) {
        v2f a, b;
        a.x = arow[k];
        a.y = arow[k + 1];
        const float* bp = bcol + (size_t)k * N;
        b.x = bp[0];
        b.y = bp[N];
        acc = __builtin_amdgcn_wmma_f32_16x16x4_f32(
            /*neg_a=*/false, a, /*neg_b=*/false, b,
            /*c_mod=*/(short)0, acc, /*reuse_a=*/false, /*reuse_b=*/false);
    }

    const int n = n0 + idx;
    const float bv = bias ? bias[n] : 0.f;
#pragma unroll
    for (int r = 0; r < 8; r++) {
        const int m = m0 + r + 8 * half;         // C layout: VGPR r -> M = r + 8*half
        const size_t o = (size_t)m * N + n;
        float v = acc[r] + bv;
        if (add0) v += add0[o];
        if (add1) v += add1[o];
        if (RELU) v = fmaxf(v, 0.f);
        C[o] = v;
    }
}

// ---------------------------------------------------------------------------
// agg[row] += x[col]  (one thread per (edge, feature); fp32 HW atomic)
// ---------------------------------------------------------------------------
__global__ void scatter_add_kernel(const float* __restrict__ x, const int* __restrict__ edges,
                                   float* __restrict__ agg, int E) {
    const int tid = blockIdx.x * blockDim.x + threadIdx.x;
    if (tid >= E * H) return;
    const int e = tid >> 7;        // /H
    const int f = tid & (H - 1);
    const int row = edges[e];
    const int col = edges[E + e];
    unsafeAtomicAdd(&agg[(size_t)row * H + f], x[(size_t)col * H + f]);
}

// h = (1+eps)*x + agg   (may run in place on agg)
__global__ void gin_combine_kernel(const float* __restrict__ x, const float* __restrict__ agg,
                                   const float* __restrict__ eps, float* __restrict__ out, int n) {
    const int i = blockIdx.x * blockDim.x + threadIdx.x;
    if (i < n) out[i] = (1.f + eps[0]) * x[i] + agg[i];
}

// x0 = x + tanh(hA*t0)+tanh(hA*t1) + tanh(hAc*t2)+tanh(hAc*t3)
__global__ void pe_kernel(const float* __restrict__ x, const float* __restrict__ hA,
                          const float* __restrict__ hAc, const float* __restrict__ theta,
                          float* __restrict__ x0, int n) {
    const int i = blockIdx.x * blockDim.x + threadIdx.x;
    if (i >= n) return;
    const int f = i & (H - 1);
    const float a = hA[i], b = hAc[i];
    x0[i] = x[i]
          + tanhf(a * theta[2 * f])           + tanhf(a * theta[2 * f + 1])
          + tanhf(b * theta[2 * (H + f)])     + tanhf(b * theta[2 * (H + f) + 1]);
}

// 32x32 LDS-tiled transpose: out[C,R] = in[R,C]^T
__global__ void transpose_kernel(const float* __restrict__ in, float* __restrict__ out,
                                 int R, int C) {
    __shared__ float tile[32][33];
    const int rb = blockIdx.y * 32, cb = blockIdx.x * 32;
    for (int s = 0; s < 32; s += 8) {
        int r = rb + threadIdx.y + s, c = cb + threadIdx.x;
        if (r < R && c < C) tile[threadIdx.y + s][threadIdx.x] = in[(size_t)r * C + c];
    }
    __syncthreads();
    for (int s = 0; s < 32; s += 8) {
        int r = cb + threadIdx.y + s, c = rb + threadIdx.x;
        if (r < C && c < R) out[(size_t)r * R + c] = tile[threadIdx.x][threadIdx.y + s];
    }
}

// e[i,j] = relu(Q_i + K_j) . wa + ba ; one wave: fixed i, 32 consecutive j
__global__ void attn_e_kernel(const float* __restrict__ Q, const float* __restrict__ Kt,
                              const float* __restrict__ wa, const float* __restrict__ ba,
                              float* __restrict__ e) {
    const int wid  = blockIdx.x * (blockDim.x >> 5) + (threadIdx.x >> 5);
    const int lane = threadIdx.x & 31;
    const int i  = wid >> 5;                 // NN/32 chunks per row
    const int j  = ((wid & 31) << 5) + lane;
    float acc = 0.f;
    const float* qi = Q + (size_t)i * H;
    for (int h = 0; h < H; h++) {
        const float r = fmaxf(qi[h] + Kt[(size_t)h * NN + j], 0.f);
        acc = fmaf(r, wa[h], acc);
    }
    e[(size_t)i * NN + j] = acc + ba[0];
}

// in-place row softmax over n columns (block of 256 per row)
__global__ void softmax_rows_kernel(float* __restrict__ e, int n) {
    float* row = e + (size_t)blockIdx.x * n;
    __shared__ float sm[8], ss[8];
    const int t = threadIdx.x, lane = t & 31, w = t >> 5;
    float m = -3.4e38f;
    for (int j = t; j < n; j += blockDim.x) m = fmaxf(m, row[j]);
    for (int off = 16; off > 0; off >>= 1) m = fmaxf(m, __shfl_down(m, off, 32));
    if (lane == 0) sm[w] = m;
    __syncthreads();
    if (t == 0) { float mm = sm[0]; for (int q = 1; q < 8; q++) mm = fmaxf(mm, sm[q]); sm[0] = mm; }
    __syncthreads();
    m = sm[0];
    float s = 0.f;
    for (int j = t; j < n; j += blockDim.x) { float v = __expf(row[j] - m); row[j] = v; s += v; }
    for (int off = 16; off > 0; off >>= 1) s += __shfl_down(s, off, 32);
    if (lane == 0) ss[w] = s;
    __syncthreads();
    if (t == 0) { float tt = 0.f; for (int q = 0; q < 8; q++) tt += ss[q]; ss[0] = tt; }
    __syncthreads();
    const float inv = 1.f / ss[0];
    for (int j = t; j < n; j += blockDim.x) row[j] *= inv;
}

// S[i,h] = sum_j alpha[i,j] * relu(Q[i,h] + K[j,h]) ; block(H) per row i
__global__ void attn_s_kernel(const float* __restrict__ Q, const float* __restrict__ Kmat,
                              const float* __restrict__ alpha, float* __restrict__ S) {
    const int i = blockIdx.x;
    const int h = threadIdx.x;
    const float q = Q[(size_t)i * H + h];
    const float* arow = alpha + (size_t)i * NN;
    float acc = 0.f;
    for (int j = 0; j < NN; j++)
        acc = fmaf(arow[j], fmaxf(q + Kmat[(size_t)j * H + h], 0.f), acc);
    S[(size_t)i * H + h] = acc;
}

// y[f] = sum_i xl[i,f]
__global__ void colsum_kernel(const float* __restrict__ xl, float* __restrict__ y) {
    const int f = threadIdx.x;
    float s = 0.f;
    for (int i = 0; i < NN; i++) s += xl[(size_t)i * H + f];
    y[f] = s;
}

// gate = sigmoid(relu(y@w1+b1)@w2+b2) ; single block of H threads
__global__ void gate_kernel(const float* __restrict__ y,
                            const float* __restrict__ w1, const float* __restrict__ b1,
                            const float* __restrict__ w2, const float* __restrict__ b2,
                            float* __restrict__ gate) {
    __shared__ float hh[H / 2];
    const int t = threadIdx.x;
    if (t < H / 2) {
        float acc = b1[t];
        for (int k = 0; k < H; k++) acc = fmaf(y[k], w1[k * (H / 2) + t], acc);
        hh[t] = fmaxf(acc, 0.f);
    }
    __syncthreads();
    if (t < H) {
        float acc = b2[t];
        for (int k = 0; k < H / 2; k++) acc = fmaf(hh[k], w2[k * H + t], acc);
        gate[t] = 1.f / (1.f + __expf(-acc));
    }
}

// xg = xl * gate[f]
__global__ void scale_kernel(const float* __restrict__ xl, const float* __restrict__ gate,
                             float* __restrict__ xg, int n) {
    const int i = blockIdx.x * blockDim.x + threadIdx.x;
    if (i < n) xg[i] = xl[i] * gate[i & (H - 1)];
}

// ---------------------------------------------------------------------------
// Host-side orchestration
// ---------------------------------------------------------------------------
extern "C" void kernel_launch(void* const* d_in, const int* in_sizes, int n_in,
                              void* d_out, int out_size, void* d_ws, size_t ws_size,
                              hipStream_t stream) {
    (void)in_sizes; (void)n_in; (void)out_size; (void)ws_size;

    // Input order: insertion order of setup_inputs(); params flattened
    // jax-tree style (keys sorted alphabetically, recursively).
    const float* x       = (const float*)d_in[0];
    const int*   eA      = (const int*)d_in[1];
    const int*   eAc     = (const int*)d_in[2];
    // d_in[3] = num_nodes (scalar, unused; constants match reference)
    const float* attn_ba  = (const float*)d_in[4];
    const float* attn_bev = (const float*)d_in[5];
    const float* attn_bk  = (const float*)d_in[6];
    const float* attn_bq  = (const float*)d_in[7];
    const float* attn_bv  = (const float*)d_in[8];
    const float* attn_wa  = (const float*)d_in[9];
    const float* attn_wev = (const float*)d_in[10];
    const float* attn_wk  = (const float*)d_in[11];
    const float* attn_wq  = (const float*)d_in[12];
    const float* attn_wv  = (const float*)d_in[13];
    const float* enca_b1  = (const float*)d_in[14];
    const float* enca_b2  = (const float*)d_in[15];
    const float* enca_eps = (const float*)d_in[16];
    const float* enca_w1  = (const float*)d_in[17];
    const float* enca_w2  = (const float*)d_in[18];
    const float* encc_b1  = (const float*)d_in[19];
    const float* encc_b2  = (const float*)d_in[20];
    const float* encc_eps = (const float*)d_in[21];
    const float* encc_w1  = (const float*)d_in[22];
    const float* encc_w2  = (const float*)d_in[23];
    const float* gi_b1    = (const float*)d_in[24];
    const float* gi_b2    = (const float*)d_in[25];
    const float* gi_w1    = (const float*)d_in[26];
    const float* gi_w2    = (const float*)d_in[27];
    const float* gina_b1  = (const float*)d_in[28];
    const float* gina_b2  = (const float*)d_in[29];
    const float* gina_eps = (const float*)d_in[30];
    const float* gina_w1  = (const float*)d_in[31];
    const float* gina_w2  = (const float*)d_in[32];
    const float* ginc_b1  = (const float*)d_in[33];
    const float* ginc_b2  = (const float*)d_in[34];
    const float* ginc_eps = (const float*)d_in[35];
    const float* ginc_w1  = (const float*)d_in[36];
    const float* ginc_w2  = (const float*)d_in[37];
    const float* out_b1   = (const float*)d_in[38];
    const float* out_b2   = (const float*)d_in[39];
    const float* out_w1   = (const float*)d_in[40];
    const float* out_w2   = (const float*)d_in[41];
    const float* theta    = (const float*)d_in[42];

    // Workspace layout (floats)
    float* w = (float*)d_ws;
    constexpr size_t NH = (size_t)NN * H;
    float* e_buf = w;                  // NN*NN (alpha in place)
    float* agg1  = e_buf + (size_t)NN * NN;
    float* agg2  = agg1 + NH;
    float* hA    = agg2 + NH;
    float* hAc   = hA + NH;
    float* tmp   = hAc + NH;
    float* x0    = tmp + NH;
    float* hA2   = x0 + NH;
    float* hsum  = hA2 + NH;
    float* Qb    = hsum + NH;
    float* Kb    = Qb + NH;
    float* Vb    = Kb + NH;
    float* Kt    = Vb + NH;
    float* Sb    = Kt + NH;
    float* attnV = Sb + NH;
    float* xl    = attnV + NH;
    float* xg    = xl + NH;
    float* o1    = xg + NH;
    float* ybuf  = o1 + NH;            // H
    float* gate  = ybuf + H;           // H

    const int EW = 256;                // elementwise block
    const int ew_blocks = (int)(NH / EW);

    auto gemm = [&](const float* A, const float* B, const float* bias,
                    const float* a0, const float* a1, float* C,
                    int M, int N, int K, bool relu) {
        const int tiles = (M >> 4) * (N >> 4);
        const int blocks = (tiles + 3) / 4;   // 4 waves / block
        if (relu)
            hipLaunchKernelGGL((wmma_gemm_f32<true>),  dim3(blocks), dim3(128), 0, stream,
                               A, B, bias, a0, a1, C, M, N, K);
        else
            hipLaunchKernelGGL((wmma_gemm_f32<false>), dim3(blocks), dim3(128), 0, stream,
                               A, B, bias, a0, a1, C, M, N, K);
    };

    auto gin = [&](const float* xin, const int* edges, const float* eps,
                   const float* w1, const float* b1, const float* w2, const float* b2,
                   float* agg, float* hid, float* dest, const float* destAdd) {
        hipMemsetAsync(agg, 0, NH * sizeof(float), stream);
        scatter_add_kernel<<<(EDGES * H) / EW, EW, 0, stream>>>(xin, edges, agg, EDGES);
        gin_combine_kernel<<<ew_blocks, EW, 0, stream>>>(xin, agg, eps, agg, (int)NH);
        gemm(agg, w1, b1, nullptr, nullptr, hid, NN, H, H, true);
        gemm(hid, w2, b2, destAdd, nullptr, dest, NN, H, H, false);
    };

    // ---- DualMPNNEncoder ----
    gin(x, eA,  enca_eps, enca_w1, enca_b1, enca_w2, enca_b2, agg1, tmp, hA,  nullptr);
    gin(x, eAc, encc_eps, encc_w1, encc_b1, encc_w2, encc_b2, agg2, tmp, hAc, nullptr);
    pe_kernel<<<ew_blocks, EW, 0, stream>>>(x, hA, hAc, theta, x0, (int)NH);

    // ---- Local layers ----
    gin(x0, eA,  gina_eps, gina_w1, gina_b1, gina_w2, gina_b2, agg1, tmp, hA2,  nullptr);
    gin(x0, eAc, ginc_eps, ginc_w1, ginc_b1, ginc_w2, ginc_b2, agg2, tmp, hsum, hA2); // hsum = hA2+hAc2

    // Attention (algebraically reduced)
    gemm(x0, attn_wq, attn_bq, nullptr, nullptr, Qb, NN, H, H, false);
    gemm(x0, attn_wk, attn_bk, nullptr, nullptr, Kb, NN, H, H, false);
    gemm(x0, attn_wv, attn_bv, nullptr, nullptr, Vb, NN, H, H, false);
    transpose_kernel<<<dim3(H / 32, NN / 32), dim3(32, 8), 0, stream>>>(Kb, Kt, NN, H);
    attn_e_kernel<<<(NN * (NN / 32)) / 8, 256, 0, stream>>>(Qb, Kt, attn_wa, attn_ba, e_buf);
    softmax_rows_kernel<<<NN, 256, 0, stream>>>(e_buf, NN);
    attn_s_kernel<<<NN, H, 0, stream>>>(Qb, Kb, e_buf, Sb);
    gemm(e_buf, Vb, nullptr, nullptr, nullptr, attnV, NN, H, NN, false); // alpha @ V
    // x_l = S@wev + bev + attnV + (hA2+hAc2)
    gemm(Sb, attn_wev, attn_bev, attnV, hsum, xl, NN, H, H, false);

    // ---- GraphInfoLayer gate ----
    colsum_kernel<<<1, H, 0, stream>>>(xl, ybuf);
    gate_kernel<<<1, H, 0, stream>>>(ybuf, gi_w1, gi_b1, gi_w2, gi_b2, gate);
    scale_kernel<<<ew_blocks, EW, 0, stream>>>(xl, gate, xg, (int)NH);

    // ---- Output MLP ----
    gemm(xg, out_w1, out_b1, nullptr, nullptr, o1, NN, H, H, true);
    gemm(o1, out_w2, out_b2, nullptr, nullptr, (float*)d_out, NN, OUT_DIM, H, false);
}